// e3LayerNorm_71554155151878
// MI455X (gfx1250) — compile-verified
//
#include <hip/hip_runtime.h>
#include <hip/hip_bf16.h>

// ---- problem constants (fixed by the reference) ----
#define TOTAL   480      // 128*1 + 64*3 + 32*5
#define NSCAL   128      // scalar (0e) features
#define NGRAPH  64
#define SPLITS  16       // blocks per graph in the reduction pass
#define PSTRIDE 608      // per-(g,s) partial slab: 480 sums + 128 sumsq
#define EPSV    1e-5f

typedef float v2f __attribute__((ext_vector_type(2)));
typedef float v8f __attribute__((ext_vector_type(8)));

// -------------------------------------------------------------------
// Kernel 1: find segment boundaries (batch is sorted int64).
// -------------------------------------------------------------------
__global__ void e3ln_init_kernel(const long long* __restrict__ batch, int N,
                                 int* __restrict__ gstart) {
    int t = threadIdx.x;
    if (t <= NGRAPH) {
        if (t == NGRAPH) {
            gstart[t] = N;
        } else {
            int lo = 0, hi = N;          // lower_bound(batch, t)
            while (lo < hi) {
                int mid = (lo + hi) >> 1;
                if (batch[mid] < (long long)t) lo = mid + 1; else hi = mid;
            }
            gstart[t] = lo;
        }
    }
}

// -------------------------------------------------------------------
// Kernel 2: per-(graph,split) partial sums via WMMA.
// Wave w owns feature tiles {w, w+8, w+16, w+24} (<30); per tile it keeps a
// per-lane pointer into column (fbase + lane%16) of row (n + lane/16).
// Main loop: 8 nodes/iter, unguarded b32 loads at immediate offsets,
// software-pipelined one stage deep, two independent WMMA accumulators:
//   D = ones(16x4) * B(4x16) + C  -> every D row = per-feature column sums.
// Sum-of-squares is accumulated unconditionally (dual-issues with WMMA) and
// only *stored* for scalar tiles -> no select chain in the loop.
// EXEC is all-ones at every WMMA (control flow is wave-uniform; the guarded
// tail restores EXEC before its WMMA).
// -------------------------------------------------------------------
__global__ void e3ln_reduce_kernel(const float* __restrict__ x,
                                   const int* __restrict__ gstart,
                                   float* __restrict__ part) {
    const int g = blockIdx.x / SPLITS;
    const int s = blockIdx.x % SPLITS;
    const int n0 = gstart[g], n1 = gstart[g + 1];
    const int glen = n1 - n0;
    const int chunk = (glen + SPLITS - 1) / SPLITS;
    const int a0 = n0 + s * chunk;
    int a1 = a0 + chunk; if (a1 > n1) a1 = n1;
    int len = (a1 > a0) ? (a1 - a0) : 0;          // slice may be empty
    const int nfull = len & ~7;                   // multiple of 8

    const int wave = threadIdx.x >> 5;   // 0..7
    const int lane = threadIdx.x & 31;
    const int half = lane >> 4;          // 0/1
    const int fcol = lane & 15;

    v2f ones; ones.x = 1.0f; ones.y = 1.0f;
    float* pbase = part + (size_t)(g * SPLITS + s) * PSTRIDE;

    for (int ft = wave; ft < 30; ft += 8) {
        const int fbase = ft * 16;
        const float* colp = x + (size_t)fbase + fcol;

        v8f acc0 = {}, acc1 = {};
        float sq0 = 0.0f, sq1 = 0.0f;

        if (nfull > 0) {
            // per-lane pointer at node (a0 + half), column fbase+fcol
            const float* p = colp + (size_t)(a0 + half) * TOTAL;
            v2f b0, b1;
            b0.x = p[0];          b0.y = p[2 * TOTAL];
            b1.x = p[4 * TOTAL];  b1.y = p[6 * TOTAL];
            p += 8 * TOTAL;
            for (int i = 8; i < nfull; i += 8) {
                v2f c0, c1;
                c0.x = p[0];          c0.y = p[2 * TOTAL];
                c1.x = p[4 * TOTAL];  c1.y = p[6 * TOTAL];
                __builtin_prefetch(p + 16 * TOTAL, 0, 3);   // global_prefetch_b8
                acc0 = __builtin_amdgcn_wmma_f32_16x16x4_f32(
                    false, ones, false, b0, (short)0, acc0, false, false);
                acc1 = __builtin_amdgcn_wmma_f32_16x16x4_f32(
                    false, ones, false, b1, (short)0, acc1, false, false);
                sq0 = fmaf(b0.x, b0.x, fmaf(b0.y, b0.y, sq0));
                sq1 = fmaf(b1.x, b1.x, fmaf(b1.y, b1.y, sq1));
                b0 = c0; b1 = c1;
                p += 8 * TOTAL;
            }
            acc0 = __builtin_amdgcn_wmma_f32_16x16x4_f32(
                false, ones, false, b0, (short)0, acc0, false, false);
            acc1 = __builtin_amdgcn_wmma_f32_16x16x4_f32(
                false, ones, false, b1, (short)0, acc1, false, false);
            sq0 = fmaf(b0.x, b0.x, fmaf(b0.y, b0.y, sq0));
            sq1 = fmaf(b1.x, b1.x, fmaf(b1.y, b1.y, sq1));
        }

        // guarded tail: at most 7 nodes (<= 2 WMMA iterations)
        for (int n = a0 + nfull; n < a1; n += 4) {
            const int na = n + half;
            const int nb = n + 2 + half;
            v2f b;
            b.x = (na < a1) ? colp[(size_t)na * TOTAL] : 0.0f;
            b.y = (nb < a1) ? colp[(size_t)nb * TOTAL] : 0.0f;
            acc0 = __builtin_amdgcn_wmma_f32_16x16x4_f32(
                false, ones, false, b, (short)0, acc0, false, false);
            sq0 = fmaf(b.x, b.x, fmaf(b.y, b.y, sq0));
        }

        // acc[0] on any lane = column sum for feature fbase + (lane%16)
        const float colsum = acc0[0] + acc1[0];
        const float sq = sq0 + sq1;
        const float sqtot = sq + __shfl_xor(sq, 16, 32);  // combine lane halves
        if (half == 0) {
            pbase[fbase + fcol] = colsum;
            if (fbase < NSCAL) pbase[480 + fbase + fcol] = sqtot;
        }
    }
}

// -------------------------------------------------------------------
// Kernel 3: fold the SPLITS partials (fixed order -> deterministic),
// compute per-graph scalar variance factor, and bake everything into
// per-(graph,feature) scale/shift:  out = x*scale + shift.
// -------------------------------------------------------------------
__global__ void e3ln_finalize_kernel(const int* __restrict__ gstart,
                                     const float* __restrict__ part,
                                     const float* __restrict__ weight,
                                     const float* __restrict__ bias,
                                     float* __restrict__ scale,
                                     float* __restrict__ shift) {
    const int g = blockIdx.x;
    const int t = threadIdx.x;              // 512 threads
    __shared__ float red[NSCAL];

    const float deg = (float)(gstart[g + 1] - gstart[g]);
    const float inv = 1.0f / (deg + 1e-12f);

    float s = 0.0f, q = 0.0f;
    if (t < TOTAL) {
        for (int sp = 0; sp < SPLITS; ++sp)
            s += part[(size_t)(g * SPLITS + sp) * PSTRIDE + t];
    }
    if (t < NSCAL) {
        for (int sp = 0; sp < SPLITS; ++sp)
            q += part[(size_t)(g * SPLITS + sp) * PSTRIDE + 480 + t];
        const float mean = s * inv;
        red[t] = q * inv - mean * mean;     // E[x^2] - mu^2 = segment variance
    }
    __syncthreads();
    for (int off = 64; off > 0; off >>= 1) {
        if (t < off) red[t] += red[t + off];
        __syncthreads();
    }
    const float norm = red[0] * (1.0f / (float)NSCAL);
    const float factor = 1.0f / (sqrtf(fmaxf(norm, 0.0f)) + EPSV);

    if (t < TOTAL) {
        const float mean = s * inv;
        float sc, sh;
        if (t < NSCAL) {                     // 0e: (x-mu)*factor*w + b
            sc = weight[t] * factor;
            sh = bias[t] - mean * sc;
        } else if (t < NSCAL + 64 * 3) {     // 1o: (x-mu)*w
            const int ch = 128 + (t - 128) / 3;
            sc = weight[ch];
            sh = -mean * sc;
        } else {                             // 2e: (x-mu)*w
            const int ch = 192 + (t - 320) / 5;
            sc = weight[ch];
            sh = -mean * sc;
        }
        scale[(size_t)g * TOTAL + t] = sc;
        shift[(size_t)g * TOTAL + t] = sh;
    }
}

// -------------------------------------------------------------------
// Kernel 4: streaming apply, float4-vectorized (480 = 120 * float4).
// scale/shift (2*120KB) stay hot in L2; this pass runs at HBM rate.
// -------------------------------------------------------------------
__global__ void e3ln_apply_kernel(const float* __restrict__ x,
                                  const long long* __restrict__ batch,
                                  const float* __restrict__ scale,
                                  const float* __restrict__ shift,
                                  float* __restrict__ out, int N) {
    const int QPR = TOTAL / 4;  // 120
    const long long idx = (long long)blockIdx.x * blockDim.x + threadIdx.x;
    const long long total = (long long)N * QPR;
    if (idx >= total) return;

    const int n = (int)(idx / QPR);
    const int q = (int)(idx % QPR);
    const int g = (int)batch[n];

    const float4 xi = ((const float4*)x)[idx];
    const float4 sc = ((const float4*)scale)[(size_t)g * QPR + q];
    const float4 sh = ((const float4*)shift)[(size_t)g * QPR + q];
    float4 o;
    o.x = fmaf(xi.x, sc.x, sh.x);
    o.y = fmaf(xi.y, sc.y, sh.y);
    o.z = fmaf(xi.z, sc.z, sh.z);
    o.w = fmaf(xi.w, sc.w, sh.w);
    ((float4*)out)[idx] = o;
}

// -------------------------------------------------------------------
extern "C" void kernel_launch(void* const* d_in, const int* in_sizes, int n_in,
                              void* d_out, int out_size, void* d_ws, size_t ws_size,
                              hipStream_t stream) {
    (void)n_in; (void)out_size; (void)ws_size;

    const float*     x      = (const float*)d_in[0];
    const float*     weight = (const float*)d_in[1];
    const float*     bias   = (const float*)d_in[2];
    const long long* batch  = (const long long*)d_in[3];
    const int N = in_sizes[3];               // 200000 nodes; NGRAPH fixed at 64

    // workspace layout (floats): [gstart pad 128][partials][scale][shift]
    float* w      = (float*)d_ws;
    int*   gstart = (int*)d_ws;                          // 65 ints, padded
    float* part   = w + 128;                             // 64*16*608 floats (~2.5MB)
    float* scale  = part + (size_t)NGRAPH * SPLITS * PSTRIDE;
    float* shift  = scale + (size_t)NGRAPH * TOTAL;
    float* out    = (float*)d_out;

    e3ln_init_kernel<<<1, 128, 0, stream>>>(batch, N, gstart);
    e3ln_reduce_kernel<<<NGRAPH * SPLITS, 256, 0, stream>>>(x, gstart, part);
    e3ln_finalize_kernel<<<NGRAPH, 512, 0, stream>>>(gstart, part, weight, bias,
                                                     scale, shift);
    const long long nvec = (long long)N * (TOTAL / 4);
    const int blocks = (int)((nvec + 255) / 256);
    e3ln_apply_kernel<<<blocks, 256, 0, stream>>>(x, batch, scale, shift, out, N);
}